// AttentiveTransformer_57870389346857
// MI455X (gfx1250) — compile-verified
//
#include <hip/hip_runtime.h>

typedef float v2f __attribute__((ext_vector_type(2)));
typedef float v8f __attribute__((ext_vector_type(8)));

#define F_DIM 512
#define D_DIM 128
#define ROWS  32
#define BN_EPS 1e-3f

__launch_bounds__(256)
__global__ void attentive_sparsemax_kernel(const float* __restrict__ inputs,
                                           const float* __restrict__ prior,
                                           const float* __restrict__ W,
                                           const float* __restrict__ bias,
                                           const float* __restrict__ gamma,
                                           const float* __restrict__ beta,
                                           const float* __restrict__ mmean,
                                           const float* __restrict__ mvar,
                                           float* __restrict__ out) {
    // Padded LDS: A stride 132 (conflict-free 16x4 f32 A-fragment b64 reads),
    // row buffer stride 516 (conflict-free tile-store and row-strided reads).
    __shared__ float sA[ROWS][D_DIM + 4];
    __shared__ float sRow[ROWS][F_DIM + 4];
    __shared__ float sScale[F_DIM];
    __shared__ float sShift[F_DIM];

    const int tid  = threadIdx.x;
    const int lane = tid & 31;
    const int wave = tid >> 5;
    const int row0 = blockIdx.x * ROWS;

    // ---- Phase 0: fold BatchNorm + dense bias into per-column scale/shift ----
    for (int c = tid; c < F_DIM; c += 256) {
        float sc = gamma[c] * rsqrtf(mvar[c] + BN_EPS);
        sScale[c] = sc;
        sShift[c] = (bias[c] - mmean[c]) * sc + beta[c];
    }

    // ---- Phase 1: stage A block (32 x 128 fp32) into LDS, float4 coalesced ----
    {
        const float4* gA = (const float4*)(inputs + (size_t)row0 * D_DIM);
        #pragma unroll
        for (int i = 0; i < (ROWS * D_DIM / 4) / 256; ++i) {
            int idx = tid + i * 256;
            int r   = idx >> 5;       // 32 float4 per row
            int c4  = (idx & 31) * 4;
            float4 v = gA[idx];
            *(float4*)(&sA[r][c4]) = v;
        }
    }
    __syncthreads();

    // ---- Phase 2: GEMM via V_WMMA_F32_16X16X4_F32 + fused BN/prior epilogue ----
    // Fragment layout (f32 16x16x4): lanes 0-15 hold K={0,1}, lanes 16-31 K={2,3}.
    const int halfk = (lane >> 4) << 1;   // 0 or 2
    const int nlan  = lane & 15;

    for (int q = 0; q < 4; ++q) {
        const int n0  = (wave * 4 + q) * 16;
        const int col = n0 + nlan;
        v8f acc0 = {};
        v8f acc1 = {};
        const float* wp = W + col;
        #pragma unroll
        for (int ks = 0; ks < D_DIM / 4; ++ks) {
            const int k = ks * 4 + halfk;
            v2f bf;
            bf.x = wp[(size_t)k       * F_DIM];
            bf.y = wp[(size_t)(k + 1) * F_DIM];
            v2f a0 = *(const v2f*)(&sA[nlan     ][k]);
            v2f a1 = *(const v2f*)(&sA[nlan + 16][k]);
            acc0 = __builtin_amdgcn_wmma_f32_16x16x4_f32(
                false, a0, false, bf, (short)0, acc0, false, false);
            acc1 = __builtin_amdgcn_wmma_f32_16x16x4_f32(
                false, a1, false, bf, (short)0, acc1, false, false);
        }
        // Epilogue: z = (gemm*scale + shift) * prior  -> LDS row buffer
        const float sc = sScale[col];
        const float sh = sShift[col];
        const int rbase = (lane >> 4) << 3;   // C/D layout: lanes>=16 hold M+8
        #pragma unroll
        for (int r = 0; r < 8; ++r) {
            int rowA = rbase + r;
            int rowB = rowA + 16;
            float p0 = __builtin_nontemporal_load(&prior[(size_t)(row0 + rowA) * F_DIM + col]);
            float p1 = __builtin_nontemporal_load(&prior[(size_t)(row0 + rowB) * F_DIM + col]);
            sRow[rowA][col] = (acc0[r] * sc + sh) * p0;
            sRow[rowB][col] = (acc1[r] * sc + sh) * p1;
        }
    }
    __syncthreads();

    // ---- Phase 3: exact sparsemax per row (Michelot iteration), 4 rows/wave ----
    for (int rr = 0; rr < 4; ++rr) {
        const int m = wave * 4 + rr;
        float z[16];
        float s = 0.0f;
        #pragma unroll
        for (int j = 0; j < 16; ++j) {
            z[j] = sRow[m][lane + 32 * j];
            s += z[j];
        }
        #pragma unroll
        for (int off = 16; off > 0; off >>= 1) s += __shfl_xor(s, off, 32);

        float tau = (s - 1.0f) * (1.0f / (float)F_DIM);
        int nsup = F_DIM;
        for (int it = 0; it < 64; ++it) {
            float ps = 0.0f;
            int   pc = 0;
            #pragma unroll
            for (int j = 0; j < 16; ++j) {
                bool in = z[j] > tau;
                ps += in ? z[j] : 0.0f;
                pc += in ? 1 : 0;
            }
            #pragma unroll
            for (int off = 16; off > 0; off >>= 1) {
                ps += __shfl_xor(ps, off, 32);
                pc += __shfl_xor(pc, off, 32);
            }
            if (pc == nsup) break;       // support unchanged -> tau is exact
            nsup = pc;
            tau  = (ps - 1.0f) / (float)pc;
        }

        float* op = out + (size_t)(row0 + m) * F_DIM;
        #pragma unroll
        for (int j = 0; j < 16; ++j) {
            __builtin_nontemporal_store(fmaxf(z[j] - tau, 0.0f), &op[lane + 32 * j]);
        }
    }
}

extern "C" void kernel_launch(void* const* d_in, const int* in_sizes, int n_in,
                              void* d_out, int out_size, void* d_ws, size_t ws_size,
                              hipStream_t stream) {
    const float* inputs = (const float*)d_in[0];
    const float* prior  = (const float*)d_in[1];
    const float* W      = (const float*)d_in[2];
    const float* bias   = (const float*)d_in[3];
    const float* gamma  = (const float*)d_in[4];
    const float* beta   = (const float*)d_in[5];
    const float* mmean  = (const float*)d_in[6];
    const float* mvar   = (const float*)d_in[7];
    float* out = (float*)d_out;

    const int Bn   = in_sizes[1] / F_DIM;   // prior is [B, F]
    const int grid = Bn / ROWS;

    attentive_sparsemax_kernel<<<grid, 256, 0, stream>>>(
        inputs, prior, W, bias, gamma, beta, mmean, mvar, out);
}